// FeedForward_decoder_82566451299223
// MI455X (gfx1250) — compile-verified
//
#include <hip/hip_runtime.h>
#include <math.h>

// ---------------------------------------------------------------------------
// FeedForward decoder for MI455X (gfx1250, wave32, WMMA f32 16x16x4 + TDM)
//   u   = W_in  (256x64)  @ x  (64 x BHW)          [WMMA GEMM, pipelined]
//   v   = dynamic per-(b,c) depthwise 3x3          [rolling-window taps]
//   g   = gelu_erf(v[:128]) * v[128:]              [VALU + LDS]
//   out = W_out (64x128)  @ g  (128 x BHW)         [WMMA GEMM, A via TDM->LDS]
// ---------------------------------------------------------------------------

typedef __attribute__((ext_vector_type(2))) float v2f;
typedef __attribute__((ext_vector_type(8))) float v8f;
typedef __attribute__((ext_vector_type(4))) unsigned int u32x4;
typedef __attribute__((ext_vector_type(8))) int i32x8;
typedef __attribute__((ext_vector_type(4))) int i32x4;

#define Bb   8
#define DIMc 64
#define C2c  256
#define HIDc 128
#define Hd   128
#define Wd   128
#define HW   16384   // Hd*Wd
#define WPAD 129     // W_sh row stride after TDM pad (1 dword per 128)

// ---------------------------------------------------------------------------
// Kernel 1: project_in as WMMA f32 GEMM. M=256 (out ch), K=64, N=HW per batch.
// Block = 256 threads = 8 waves; wave (mg,ng) -> 64 M rows x 16 pixels.
// fp32 16x16x4 layout: lanes 0..15 hold K=k0+0/1, lanes 16..31 hold K=k0+2/3.
// ---------------------------------------------------------------------------
__global__ __launch_bounds__(256) void proj_in_wmma(
    const float* __restrict__ x, const float* __restrict__ W_in,
    float* __restrict__ u)
{
    const int b    = blockIdx.x >> 9;        // / 512 pixel tiles
    const int tile = blockIdx.x & 511;
    const int wave = threadIdx.x >> 5;
    const int lane = threadIdx.x & 31;
    const int half = lane >> 4;
    const int l15  = lane & 15;
    const int mg   = wave >> 1;              // 0..3 -> channel group of 64
    const int ng   = wave & 1;               // 0..1 -> pixel group of 16
    const int pcol = tile * 32 + ng * 16 + l15;

    v8f acc0 = {}, acc1 = {}, acc2 = {}, acc3 = {};

    const float* __restrict__ xb = x + (size_t)b * DIMc * HW + pcol;

    for (int k0 = 0; k0 < DIMc; k0 += 4) {
        const int ka = k0 + 2 * half;        // this lane-half's K pair
        v2f bb;
        bb.x = xb[(size_t)ka * HW];
        bb.y = xb[(size_t)(ka + 1) * HW];
        if (k0 + 4 < DIMc)
            __builtin_prefetch(xb + (size_t)(ka + 4) * HW, 0, 0);

        const float* __restrict__ wrow = W_in + (mg * 64 + l15) * DIMc + ka;
        v2f a0, a1, a2, a3;
        a0.x = wrow[0];            a0.y = wrow[1];
        a1.x = wrow[16 * DIMc];    a1.y = wrow[16 * DIMc + 1];
        a2.x = wrow[32 * DIMc];    a2.y = wrow[32 * DIMc + 1];
        a3.x = wrow[48 * DIMc];    a3.y = wrow[48 * DIMc + 1];

        acc0 = __builtin_amdgcn_wmma_f32_16x16x4_f32(false, a0, false, bb, (short)0, acc0, false, false);
        acc1 = __builtin_amdgcn_wmma_f32_16x16x4_f32(false, a1, false, bb, (short)0, acc1, false, false);
        acc2 = __builtin_amdgcn_wmma_f32_16x16x4_f32(false, a2, false, bb, (short)0, acc2, false, false);
        acc3 = __builtin_amdgcn_wmma_f32_16x16x4_f32(false, a3, false, bb, (short)0, acc3, false, false);
    }

    float* __restrict__ ub = u + (size_t)b * C2c * HW + pcol;
#pragma unroll
    for (int v = 0; v < 8; ++v) {
        const int mrow = v + 8 * half;
        ub[(size_t)(mg * 64 +  0 + mrow) * HW] = acc0[v];
        ub[(size_t)(mg * 64 + 16 + mrow) * HW] = acc1[v];
        ub[(size_t)(mg * 64 + 32 + mrow) * HW] = acc2[v];
        ub[(size_t)(mg * 64 + 48 + mrow) * HW] = acc3[v];
    }
}

// ---------------------------------------------------------------------------
// Kernel 2: fused dynamic-depthwise 3x3 + exact-erf GELU gate + project_out.
// Block = 256 threads handles (b, row y, 32-col segment).
//   Wave 0 issues a TENSOR_LOAD_TO_LDS of W_out (64x128 f32, 2D D#) with
//   1-dword-per-row LDS padding (row stride 129 -> bank-conflict-free A reads)
//   overlapping the conv phase; s_wait_tensorcnt 0 + barrier before GEMM.
// ---------------------------------------------------------------------------
__global__ __launch_bounds__(256) void dw_gate_proj_out(
    const float* __restrict__ u, const float* __restrict__ gen_kernel,
    const float* __restrict__ dw_kernel, const float* __restrict__ lambda_dw,
    const float* __restrict__ W_out, float* __restrict__ out)
{
    __shared__ float v_sh[C2c][33];       // 256 x 33 (stride-33: conflict-free)
    __shared__ float g_sh[HIDc][33];      // 128 x 33
    __shared__ float W_sh[DIMc * WPAD];   // 64 x 129 (TDM-padded rows)

    const int b   = blockIdx.x / (Hd * 4);
    const int rem = blockIdx.x % (Hd * 4);
    const int y   = rem >> 2;
    const int seg = rem & 3;
    const int x0  = seg * 32;
    const int t   = threadIdx.x;

    // ---- Issue TDM: DMA W_out (64x128 f32) into W_sh, padded rows ----
    if (t < 32) {
        const unsigned lds_addr = (unsigned)(size_t)(&W_sh[0]);
        const unsigned long long ga = (unsigned long long)(size_t)W_out;
        u32x4 g0;
        g0[0] = 1u;                                    // count=1 valid descriptor
        g0[1] = lds_addr;                              // LDS byte address
        g0[2] = (unsigned)(ga & 0xFFFFFFFFu);          // global_addr[31:0]
        g0[3] = (unsigned)((ga >> 32) & 0x1FFFFFFu)    // global_addr[56:32]
              | (2u << 30);                            // type=2 ("image")
        i32x8 g1;
        g1[0] = (2 << 16)        // data_size = 4 bytes
              | (1 << 20)        // pad_enable
              | (6 << 22);       // pad_interval: 128 dwords
                                 // pad_amount = 0 -> 1 dword pad per row
        g1[1] = (HIDc << 16);    // tensor_dim0 = 128 (bits 79:48 lo half)
        g1[2] = (DIMc << 16);    // tensor_dim1 = 64  (bits 111:80 lo half)
        g1[3] = (HIDc << 16);    // tile_dim0 = 128
        g1[4] = DIMc;            // tile_dim1 = 64, tile_dim2 = 0
        g1[5] = HIDc;            // tensor_dim0_stride = 128
        g1[6] = 0;               // stride0 hi, stride1 lo
        g1[7] = 0;               // stride1 hi
        i32x4 g2 = {0, 0, 0, 0};                 // unused (2D tensor)
        i32x4 g3 = {0, 0, 0, 0};
        i32x8 g4 = {0, 0, 0, 0, 0, 0, 0, 0};     // extra group (6-arg toolchain)
        __builtin_amdgcn_tensor_load_to_lds(g0, g1, g2, g3, g4, 0);
    }

    // ---- Phase 1: dynamic depthwise 3x3, rolling 3-column window ----
    {
        const int c = t;                 // one channel per thread
        const float lam = lambda_dw[c];
        const float* __restrict__ dwp = dw_kernel + c * 9;
        const float* __restrict__ gkp = gen_kernel + ((size_t)b * C2c + c) * 9;
        float kk[9];
#pragma unroll
        for (int j = 0; j < 9; ++j) kk[j] = dwp[j] + lam * gkp[j];

        const float* __restrict__ up = u + ((size_t)b * C2c + c) * HW;
        const bool ym_ok = (y > 0), yp_ok = (y < Hd - 1);
        const float* __restrict__ r0 = up + (y - 1) * Wd;
        const float* __restrict__ r1 = up + y * Wd;
        const float* __restrict__ r2 = up + (y + 1) * Wd;

        float cL0, cL1, cL2, cC0, cC1, cC2;   // left / center columns (3 rows)
        {
            const int xl = x0 - 1;
            const bool xok = (xl >= 0);
            cL0 = (xok && ym_ok) ? r0[xl] : 0.0f;
            cL1 =  xok           ? r1[xl] : 0.0f;
            cL2 = (xok && yp_ok) ? r2[xl] : 0.0f;
            cC0 = ym_ok ? r0[x0] : 0.0f;
            cC1 =         r1[x0];
            cC2 = yp_ok ? r2[x0] : 0.0f;
        }
        for (int px = 0; px < 32; ++px) {
            const int xr = x0 + px + 1;
            const bool xok = (xr < Wd);
            const float cR0 = (xok && ym_ok) ? r0[xr] : 0.0f;
            const float cR1 =  xok           ? r1[xr] : 0.0f;
            const float cR2 = (xok && yp_ok) ? r2[xr] : 0.0f;
            const float s = kk[0] * cL0 + kk[1] * cC0 + kk[2] * cR0
                          + kk[3] * cL1 + kk[4] * cC1 + kk[5] * cR1
                          + kk[6] * cL2 + kk[7] * cC2 + kk[8] * cR2;
            v_sh[c][px] = s;
            cL0 = cC0; cL1 = cC1; cL2 = cC2;
            cC0 = cR0; cC1 = cR1; cC2 = cR2;
        }
    }

    // TDM must have landed before anyone reads W_sh after the barrier
    if (t < 32) __builtin_amdgcn_s_wait_tensorcnt(0);
    __syncthreads();

    // ---- Phase 2: exact-erf GELU gating ----
    {
        const int c     = t & 127;
        const int pbase = (t >> 7) * 16;
#pragma unroll
        for (int i = 0; i < 16; ++i) {
            const float a  = v_sh[c][pbase + i];
            const float ge = 0.5f * a * (1.0f + erff(a * 0.70710678118654752f));
            g_sh[c][pbase + i] = ge * v_sh[c + HIDc][pbase + i];
        }
    }
    __syncthreads();

    // ---- Phase 3: project_out WMMA GEMM, M=64 K=128 N=32, A+B from LDS ----
    const int wave = t >> 5, lane = t & 31;
    const int half = lane >> 4, l15 = lane & 15;
    const int m0 = (wave >> 1) * 16;   // 4 M-tiles
    const int n0 = (wave & 1) * 16;    // 2 N-tiles

    v8f acc = {};
    const float* __restrict__ wrow = &W_sh[(m0 + l15) * WPAD];
    for (int k0 = 0; k0 < HIDc; k0 += 4) {
        const int ka = k0 + 2 * half;
        v2f aa, bb;
        aa.x = wrow[ka];                  // ds_load, stride-129 rows: no conflicts
        aa.y = wrow[ka + 1];
        bb.x = g_sh[ka][n0 + l15];
        bb.y = g_sh[ka + 1][n0 + l15];
        acc = __builtin_amdgcn_wmma_f32_16x16x4_f32(false, aa, false, bb, (short)0, acc, false, false);
    }

    float* __restrict__ op = out + (size_t)b * DIMc * HW + y * Wd + x0 + n0 + l15;
#pragma unroll
    for (int v = 0; v < 8; ++v) {
        op[(size_t)(m0 + v + 8 * half) * HW] = acc[v];
    }
}

// ---------------------------------------------------------------------------
extern "C" void kernel_launch(void* const* d_in, const int* in_sizes, int n_in,
                              void* d_out, int out_size, void* d_ws, size_t ws_size,
                              hipStream_t stream)
{
    const float* x     = (const float*)d_in[0];  // (8,64,128,128)
    const float* gen_k = (const float*)d_in[1];  // (8,256,1,3,3)
    const float* W_in  = (const float*)d_in[2];  // (256,64)
    const float* dw_k  = (const float*)d_in[3];  // (1,256,1,3,3)
    const float* lam   = (const float*)d_in[4];  // (1,256,1,1,1)
    const float* W_out = (const float*)d_in[5];  // (64,128)
    float*       out   = (float*)d_out;          // (8,64,128,128)

    float* u = (float*)d_ws;                     // (8,256,128,128) = 134 MB

    dim3 blk(256);
    proj_in_wmma<<<dim3(Bb * 512), blk, 0, stream>>>(x, W_in, u);
    dw_gate_proj_out<<<dim3(Bb * Hd * 4), blk, 0, stream>>>(u, gen_k, dw_k, lam, W_out, out);
}